// PhysNetEdgeEmbeddingBlock_20272245637563
// MI455X (gfx1250) — compile-verified
//
#include <hip/hip_runtime.h>
#include <hip/hip_bf16.h>

typedef __attribute__((ext_vector_type(2))) float v2f;
typedef __attribute__((ext_vector_type(8))) float v8f;

#define CUTOFF 5.0f

__device__ __forceinline__ v8f wmma4(v2f a, v2f b, v8f c) {
  // D = A(16x4) * B(4x16) + C, f32, wave32
  return __builtin_amdgcn_wmma_f32_16x16x4_f32(
      /*neg_a=*/false, a, /*neg_b=*/false, b,
      /*c_mod=*/(short)0, c, /*reuse_a=*/false, /*reuse_b=*/false);
}

// Main kernel: full 32-edge tiles only. One wave = 32 edges; lane l does the
// scalar work for edge base+l, then the wave evaluates
//   exponent[e][k] = [x^2, x, 1] . [-beta_k, 2 beta_k mu_k, -beta_k mu_k^2]
// for the 32x32 tile with four V_WMMA_F32_16X16X4_F32 ops. No guards anywhere.
__global__ __launch_bounds__(256) void physnet_rbf_wmma_kernel(
    const float* __restrict__ coords,      // [N,3]
    const long long* __restrict__ recv,    // [E] int64
    const long long* __restrict__ send,    // [E] int64
    const float* __restrict__ mu,          // [32]
    const float* __restrict__ beta,        // [32]
    float* __restrict__ out,               // [E,32]
    int nFullWaves)                        // nEdges / 32
{
  const int  lane = threadIdx.x & 31;
  const bool lo   = lane < 16;
  const int  n0   = lane & 15;

  // readfirstlane -> SGPR: base & bounds check become scalar (s_cbranch, no exec dance)
  const int waveId = blockIdx.x * 8 + __builtin_amdgcn_readfirstlane(threadIdx.x >> 5);
  if (waveId >= nFullWaves) return;       // uniform scalar branch
  const long long base = (long long)waveId * 32;

  // ---- B tiles (4x16 f32, KxN). ISA layout: VGPR0 = {K=0 | K=2}, VGPR1 = {K=1 | K=3}
  //      rows: K0 = -beta, K1 = 2*beta*mu, K2 = -beta*mu^2, K3 = 0
  const float mA = mu[n0],      bA = beta[n0];
  const float mB = mu[n0 + 16], bB = beta[n0 + 16];
  v2f Blo, Bhi;
  Blo.x = lo ? -bA        : -bA * mA * mA;
  Blo.y = lo ? 2.0f*bA*mA : 0.0f;
  Bhi.x = lo ? -bB        : -bB * mB * mB;
  Bhi.y = lo ? 2.0f*bB*mB : 0.0f;

  // ---- per-edge scalar work (one edge per lane, always in range)
  const long long e  = base + lane;
  const long long ri = recv[e];
  const long long si = send[e];
  const float dx = coords[ri*3 + 0] - coords[si*3 + 0];
  const float dy = coords[ri*3 + 1] - coords[si*3 + 1];
  const float dz = coords[ri*3 + 2] - coords[si*3 + 2];
  const float r  = sqrtf(dx*dx + dy*dy + dz*dz);
  const float u  = r * (1.0f / CUTOFF);
  const float u3 = u * u * u;
  const float phi = 1.0f + u3 * ((-6.0f*u + 15.0f)*u - 10.0f); // 1 - 6u^5 + 15u^4 - 10u^3
  const float x  = __expf(-r);
  const float x2 = x * x;

  // ---- A tiles (16x4 f32, MxK). Lane=M, VGPR0 = {K=0 | K=2}, VGPR1 = {K=1 | K=3}
  //      row m: [x2_m, x_m, 1, 0]; upper 16 edges pulled down via shuffle
  const float x2h = __shfl(x2, lane | 16);
  const float xh  = __shfl(x,  lane | 16);
  v2f Alo, Ahi;
  Alo.x = lo ? x2  : 1.0f;
  Alo.y = lo ? x   : 0.0f;
  Ahi.x = lo ? x2h : 1.0f;
  Ahi.y = lo ? xh  : 0.0f;

  const v8f c0 = {};
  v8f D[2][2];
  D[0][0] = wmma4(Alo, Blo, c0);  // edges 0-15  x basis 0-15
  D[0][1] = wmma4(Alo, Bhi, c0);  // edges 0-15  x basis 16-31
  D[1][0] = wmma4(Ahi, Blo, c0);  // edges 16-31 x basis 0-15
  D[1][1] = wmma4(Ahi, Bhi, c0);  // edges 16-31 x basis 16-31

  // ---- epilogue: rbf = exp(exponent) * phi(row); straight-line, no guards
  //      D layout: VGPR v -> M = v (lanes 0-15) / v+8 (lanes 16-31)
  float* obase = out + base * 32;                 // scalar address part
  float* o     = obase + (lo ? 0 : 8) * 32 + n0;  // + per-lane offset
  #pragma unroll
  for (int eh = 0; eh < 2; ++eh) {
    #pragma unroll
    for (int v = 0; v < 8; ++v) {
      const float ph  = __shfl(phi, (lo ? v : v + 8) + eh * 16);
      const int   off = (eh * 16 + v) * 32;       // constant byte offsets in stores
      __builtin_nontemporal_store(__expf(D[eh][0][v]) * ph, o + off);
      __builtin_nontemporal_store(__expf(D[eh][1][v]) * ph, o + off + 16);
    }
  }
}

// Tail kernel: handles nEdges % 32 leftover edges, one (edge, basis) element
// per thread. Tiny, launched at most once with a single block.
__global__ void physnet_rbf_tail_kernel(
    const float* __restrict__ coords,
    const long long* __restrict__ recv,
    const long long* __restrict__ send,
    const float* __restrict__ mu,
    const float* __restrict__ beta,
    float* __restrict__ out,
    int nEdges, int tailStart)
{
  const int tid  = blockIdx.x * blockDim.x + threadIdx.x;
  const int edge = tailStart + (tid >> 5);
  const int k    = tid & 31;
  if (edge >= nEdges) return;
  const long long ri = recv[edge];
  const long long si = send[edge];
  const float dx = coords[ri*3 + 0] - coords[si*3 + 0];
  const float dy = coords[ri*3 + 1] - coords[si*3 + 1];
  const float dz = coords[ri*3 + 2] - coords[si*3 + 2];
  const float r  = sqrtf(dx*dx + dy*dy + dz*dz);
  const float u  = r * (1.0f / CUTOFF);
  const float u3 = u * u * u;
  const float phi = 1.0f + u3 * ((-6.0f*u + 15.0f)*u - 10.0f);
  const float diff = __expf(-r) - mu[k];
  out[(long long)edge * 32 + k] = __expf(-beta[k] * diff * diff) * phi;
}

extern "C" void kernel_launch(void* const* d_in, const int* in_sizes, int n_in,
                              void* d_out, int out_size, void* d_ws, size_t ws_size,
                              hipStream_t stream) {
  const float*     coords = (const float*)d_in[0];      // [100000,3] f32
  const long long* recv   = (const long long*)d_in[1];  // [1,2000000] i64
  const long long* send   = (const long long*)d_in[2];  // [1,2000000] i64
  const float*     mu     = (const float*)d_in[3];      // [1,32] f32
  const float*     beta   = (const float*)d_in[4];      // [1,32] f32
  float*           out    = (float*)d_out;              // [2000000,32] f32

  const int nEdges     = in_sizes[1];
  const int nFullWaves = nEdges / 32;                   // full 32-edge tiles
  const int tailStart  = nFullWaves * 32;

  if (nFullWaves > 0) {
    const int blocks = (nFullWaves + 7) / 8;            // 8 waves (256 thr) per block
    physnet_rbf_wmma_kernel<<<blocks, 256, 0, stream>>>(
        coords, recv, send, mu, beta, out, nFullWaves);
  }
  if (tailStart < nEdges) {                             // never taken for 2M edges
    physnet_rbf_tail_kernel<<<1, 1024, 0, stream>>>(
        coords, recv, send, mu, beta, out, nEdges, tailStart);
  }
}